// GraphConvNet_2602750181798
// MI455X (gfx1250) — compile-verified
//
#include <hip/hip_runtime.h>

typedef __attribute__((ext_vector_type(2))) float v2f;
typedef __attribute__((ext_vector_type(8))) float v8f;

#define LATENT 128
#define F_IN 16
#define STEPS 3
#define OUT_GLOBALS 10
#define LN_EPS 1e-6f

// ---------------- degree / normalization ----------------
__global__ void deg_init_kernel(float* ds, float* dr, int n) {
  int i = blockIdx.x * blockDim.x + threadIdx.x;
  if (i < n) { ds[i] = 1.0f; dr[i] = 1.0f; }  // self-edge contributes 1 to each degree
}

__global__ void deg_accum_kernel(const int* __restrict__ snd, const int* __restrict__ rcv,
                                 float* ds, float* dr, int e) {
  int i = blockIdx.x * blockDim.x + threadIdx.x;
  if (i < e) {
    atomicAdd(&ds[snd[i]], 1.0f);
    atomicAdd(&dr[rcv[i]], 1.0f);
  }
}

__global__ void deg_finish_kernel(float* ds, float* dr, int n) {
  int i = blockIdx.x * blockDim.x + threadIdx.x;
  if (i < n) {
    ds[i] = rsqrtf(fmaxf(ds[i], 1.0f));
    dr[i] = rsqrtf(fmaxf(dr[i], 1.0f));
  }
}

// ---------------- fp32 WMMA GEMM: out[N,128] = act(A[N,K] @ W[K,128] + b) ----------------
// Each block: 256 threads = 8 waves, each wave computes a 16-row x 128-col stripe.
// W is staged in LDS swizzled so each B fragment (16x16x4 tile, K-pair) is a contiguous b64.
template<int K_DIM, bool RELU, bool SCALE, bool DUP>
__global__ __launch_bounds__(256) void gemm_node(
    const float* __restrict__ A, const float* __restrict__ W,
    const float* __restrict__ bias, const float* __restrict__ rowscale,
    float* __restrict__ out, float* __restrict__ out2, int nrows)
{
  __shared__ float wl[K_DIM * LATENT];
  const int tid = threadIdx.x;
  for (int i = tid; i < K_DIM * LATENT; i += 256) {
    const int k = i >> 7;         // row of W
    const int n = i & 127;        // col of W
    // swizzle: pair (k, k+1) adjacent so B fragments load as aligned float2
    wl[(k >> 1) * 256 + (n << 1) + (k & 1)] = W[i];
  }
  __syncthreads();

  const int wave = tid >> 5;
  const int lane = tid & 31;
  const int hf   = lane >> 4;     // 0: K0..1 / M,  1: K2..3 / M+8 halves per ISA layout
  const int lid  = lane & 15;
  const int m0   = (blockIdx.x * 8 + wave) * 16;
  if (m0 >= nrows) return;

  v8f acc[8] = {};                // 8 tiles of 16x16 cover N=128

  for (int kb = 0; kb < K_DIM; kb += 4) {
    // A fragment: 16x4, lane<16 holds {K=kb,kb+1} of row m0+lid, lane>=16 holds {kb+2,kb+3}
    const float* ap = A + (size_t)(m0 + lid) * K_DIM + (kb + 2 * hf);
    const v2f a = *(const v2f*)ap;
    // B fragments from swizzled LDS
    const float* bp = &wl[((kb >> 1) + hf) * 256 + (lid << 1)];
#pragma unroll
    for (int t = 0; t < 8; ++t) {
      const v2f b = *(const v2f*)(bp + t * 32);
      acc[t] = __builtin_amdgcn_wmma_f32_16x16x4_f32(
          false, a, false, b, (short)0, acc[t], false, false);
    }
  }

  float bn[8];
#pragma unroll
  for (int t = 0; t < 8; ++t) bn[t] = bias[t * 16 + lid];

#pragma unroll
  for (int v = 0; v < 8; ++v) {
    const int m = m0 + v + 8 * hf;          // C/D layout: VGPR v -> row v (+8 for upper half)
    float sc = 1.0f;
    if (SCALE) sc = rowscale[m];
#pragma unroll
    for (int t = 0; t < 8; ++t) {
      const int n = t * 16 + lid;
      float val = acc[t][v] + bn[t];
      if (RELU) val = fmaxf(val, 0.0f);
      if (SCALE) val *= sc;
      out[(size_t)m * LATENT + n] = val;
      if (DUP) out2[(size_t)m * LATENT + n] = val;   // agg initialized with self-edge term
    }
  }
}

// ---------------- edge scatter: agg[r] += m[s], one wave per edge ----------------
__global__ void edge_scatter_kernel(const float* __restrict__ m, const int* __restrict__ snd,
                                    const int* __restrict__ rcv, float* __restrict__ agg, int ne) {
  const int idx = blockIdx.x * blockDim.x + threadIdx.x;
  const int e = idx >> 5;
  if (e >= ne) return;
  const int c = (idx & 31) << 2;
  const int s = snd[e];
  const int r = rcv[e];
  const float4 v = *(const float4*)(m + (size_t)s * LATENT + c);
  float* dst = agg + (size_t)r * LATENT + c;
  atomicAdd(dst + 0, v.x);
  atomicAdd(dst + 1, v.y);
  atomicAdd(dst + 2, v.z);
  atomicAdd(dst + 3, v.w);
}

// ---------------- h = LayerNorm(agg*inv_r + h), one wave per node ----------------
__global__ void skip_ln_kernel(const float* __restrict__ agg, const float* __restrict__ invr,
                               float* __restrict__ h, const float* __restrict__ gamma,
                               const float* __restrict__ beta, int n) {
  const int node = (blockIdx.x * blockDim.x + threadIdx.x) >> 5;
  const int lane = threadIdx.x & 31;
  if (node >= n) return;
  const size_t base = (size_t)node * LATENT + (lane << 2);
  const float4 a = *(const float4*)(agg + base);
  float4 x = *(const float4*)(h + base);
  const float ir = invr[node];
  x.x = fmaf(a.x, ir, x.x);
  x.y = fmaf(a.y, ir, x.y);
  x.z = fmaf(a.z, ir, x.z);
  x.w = fmaf(a.w, ir, x.w);

  float s = x.x + x.y + x.z + x.w;
#pragma unroll
  for (int o = 16; o >= 1; o >>= 1) s += __shfl_xor(s, o, 32);
  const float mu = s * (1.0f / 128.0f);

  const float dx = x.x - mu, dy = x.y - mu, dz = x.z - mu, dw = x.w - mu;
  float q = dx * dx + dy * dy + dz * dz + dw * dw;
#pragma unroll
  for (int o = 16; o >= 1; o >>= 1) q += __shfl_xor(q, o, 32);
  const float rs = rsqrtf(q * (1.0f / 128.0f) + LN_EPS);

  const float4 g = *(const float4*)(gamma + (lane << 2));
  const float4 b = *(const float4*)(beta + (lane << 2));
  float4 o4;
  o4.x = fmaf(dx * rs, g.x, b.x);
  o4.y = fmaf(dy * rs, g.y, b.y);
  o4.z = fmaf(dz * rs, g.z, b.z);
  o4.w = fmaf(dw * rs, g.w, b.w);
  *(float4*)(h + base) = o4;
}

// ---------------- pooling + decoder ----------------
__global__ void pool_zero_kernel(float* pooled, float* counts, int g) {
  const int i = blockIdx.x * blockDim.x + threadIdx.x;
  if (i < g * LATENT) pooled[i] = 0.0f;
  if (i < g) counts[i] = 0.0f;
}

__global__ void pool_accum_kernel(const float* __restrict__ h, const int* __restrict__ gid,
                                  float* pooled, float* counts, int n) {
  const int idx = blockIdx.x * blockDim.x + threadIdx.x;
  const int node = idx >> 5;
  if (node >= n) return;
  const int c = (idx & 31) << 2;
  const int g = gid[node];
  const float4 v = *(const float4*)(h + (size_t)node * LATENT + c);
  float* dst = pooled + (size_t)g * LATENT + c;
  atomicAdd(dst + 0, v.x);
  atomicAdd(dst + 1, v.y);
  atomicAdd(dst + 2, v.z);
  atomicAdd(dst + 3, v.w);
  if (c == 0) atomicAdd(&counts[g], 1.0f);
}

__global__ void decode_kernel(const float* __restrict__ pooled, const float* __restrict__ counts,
                              const float* __restrict__ W, const float* __restrict__ b,
                              float* __restrict__ out, int g) {
  const int idx = blockIdx.x * blockDim.x + threadIdx.x;
  if (idx >= g * OUT_GLOBALS) return;
  const int gi = idx / OUT_GLOBALS;
  const int o = idx % OUT_GLOBALS;
  const float inv = 1.0f / fmaxf(counts[gi], 1.0f);
  const float* p = pooled + (size_t)gi * LATENT;
  float acc = b[o];
  for (int c = 0; c < LATENT; ++c)
    acc = fmaf(p[c] * inv, W[c * OUT_GLOBALS + o], acc);
  out[idx] = acc;
}

// ---------------- host launcher ----------------
extern "C" void kernel_launch(void* const* d_in, const int* in_sizes, int n_in,
                              void* d_out, int out_size, void* d_ws, size_t ws_size,
                              hipStream_t stream) {
  const float* nodes   = (const float*)d_in[0];
  const int*   senders = (const int*)d_in[1];
  const int*   recv    = (const int*)d_in[2];
  const int*   gids    = (const int*)d_in[3];
  const float* embW    = (const float*)d_in[4];
  const float* embB    = (const float*)d_in[5];
  const float* mlpW    = (const float*)d_in[6];
  const float* mlpB    = (const float*)d_in[7];
  const float* lnS     = (const float*)d_in[8];
  const float* lnB     = (const float*)d_in[9];
  const float* decW    = (const float*)d_in[10];
  const float* decB    = (const float*)d_in[11];
  float* out = (float*)d_out;

  const int N = in_sizes[0] / F_IN;
  const int E = in_sizes[1];
  const int G = out_size / OUT_GLOBALS;
  (void)n_in; (void)ws_size;

  size_t off = 0;
  auto carve = [&](size_t bytes) -> float* {
    float* p = (float*)((char*)d_ws + off);
    off += (bytes + 255) & ~(size_t)255;
    return p;
  };
  float* h      = carve((size_t)N * LATENT * sizeof(float));
  float* m      = carve((size_t)N * LATENT * sizeof(float));
  float* agg    = carve((size_t)N * LATENT * sizeof(float));
  float* inv_s  = carve((size_t)N * sizeof(float));
  float* inv_r  = carve((size_t)N * sizeof(float));
  float* pooled = carve((size_t)G * LATENT * sizeof(float));
  float* counts = carve((size_t)G * sizeof(float));

  // degrees -> D^{-1/2}
  deg_init_kernel<<<(N + 255) / 256, 256, 0, stream>>>(inv_s, inv_r, N);
  deg_accum_kernel<<<(E + 255) / 256, 256, 0, stream>>>(senders, recv, inv_s, inv_r, E);
  deg_finish_kernel<<<(N + 255) / 256, 256, 0, stream>>>(inv_s, inv_r, N);

  const int gemm_blocks = (N + 127) / 128;  // 8 waves x 16 rows per block

  // embedder: h = nodes @ embW + embB (no activation)
  gemm_node<F_IN, false, false, false><<<gemm_blocks, 256, 0, stream>>>(
      nodes, embW, embB, nullptr, h, nullptr, N);

  for (int s = 0; s < STEPS; ++s) {
    const float* W0 = mlpW + (size_t)s * 2 * LATENT * LATENT;
    const float* W1 = W0 + (size_t)LATENT * LATENT;
    const float* b0 = mlpB + (size_t)s * 2 * LATENT;
    const float* b1 = b0 + LATENT;

    // m = relu(h @ W0 + b0)
    gemm_node<LATENT, true, false, false><<<gemm_blocks, 256, 0, stream>>>(
        h, W0, b0, nullptr, m, nullptr, N);
    // m = relu(m @ W1 + b1) * inv_s ; agg initialized to m (self edge)
    gemm_node<LATENT, true, true, true><<<gemm_blocks, 256, 0, stream>>>(
        m, W1, b1, inv_s, m, agg, N);

    // agg[r] += m[snd] over edges
    const long sc_threads = (long)E * 32;
    edge_scatter_kernel<<<(int)((sc_threads + 255) / 256), 256, 0, stream>>>(
        m, senders, recv, agg, E);

    // h = LN(agg * inv_r + h)
    const long ln_threads = (long)N * 32;
    skip_ln_kernel<<<(int)((ln_threads + 255) / 256), 256, 0, stream>>>(
        agg, inv_r, h, lnS + (size_t)s * LATENT, lnB + (size_t)s * LATENT, N);
  }

  // segment-mean pool + decoder
  pool_zero_kernel<<<(G * LATENT + 255) / 256, 256, 0, stream>>>(pooled, counts, G);
  const long pl_threads = (long)N * 32;
  pool_accum_kernel<<<(int)((pl_threads + 255) / 256), 256, 0, stream>>>(
      h, gids, pooled, counts, N);
  decode_kernel<<<(G * OUT_GLOBALS + 255) / 256, 256, 0, stream>>>(
      pooled, counts, decW, decB, out, G);
}